// MPNEncoder_77653008711810
// MI455X (gfx1250) — compile-verified
//
#include <hip/hip_runtime.h>
#include <hip/hip_bf16.h>

typedef __attribute__((ext_vector_type(2))) float v2f;
typedef __attribute__((ext_vector_type(8))) float v8f;

#define HID 300
#define HQ 75            // HID / 4 (float4 chunks)
#define ATOMF 133
#define BONDF 147
#define MAXNB 6
#define BN_EPS 1e-5f

#define TILE_M 128
#define TILE_N 64
#define TILE_K 64
#define A_LDS_STRIDE 68  // 68 mod 64 == 4 -> 16 rows hit 16 distinct banks
#define W_LDS_STRIDE 80  // 2*80 mod 64 == 32 -> upper half-wave hits disjoint bank range

// ---------------------------------------------------------------------------
// Epilogue helper: one 16x16 C/D fragment. VGPR i: lanes0-15 -> M=i, lanes16-31 -> M=i+8
// ---------------------------------------------------------------------------
__device__ __forceinline__ void epilogue_store(const v8f& c, int row0, int col,
                                               int M, int N,
                                               const float* addv, const float* bias,
                                               int do_relu, float* D, float* D2) {
    if (col >= N) return;
    float bv = bias ? bias[col] : 0.0f;
#pragma unroll
    for (int i = 0; i < 8; ++i) {
        int row = row0 + i;
        if (row < M) {
            float v = c[i] + bv;
            if (addv) v += addv[(long)row * N + col];
            if (D2) D2[(long)row * N + col] = v;
            D[(long)row * N + col] = do_relu ? fmaxf(v, 0.0f) : v;
        }
    }
}

// ---------------------------------------------------------------------------
// Generic fp32 WMMA GEMM:  D = act( [A0 | A1] @ W + addv + bias )
//   A0: [M, K0], A1: [M, K1] (nullable), W: [(K0+K1), N], addv: [M,N] (nullable)
//   D2 (nullable): pre-activation output
// ---------------------------------------------------------------------------
__launch_bounds__(256)
__global__ void k_gemm_wmma_f32(const float* __restrict__ A0, int K0,
                                const float* __restrict__ A1, int K1,
                                const float* __restrict__ W,
                                const float* __restrict__ addv,
                                const float* __restrict__ bias,
                                float* __restrict__ D,
                                float* __restrict__ D2,
                                int M, int N, int do_relu) {
    __shared__ float Alds[TILE_M * A_LDS_STRIDE];
    __shared__ float Wlds[TILE_K * W_LDS_STRIDE];

    const int Ktot = K0 + K1;
    const int tid  = threadIdx.x;
    const int lane = tid & 31;
    const int wv   = tid >> 5;                 // 8 waves per block
    const int rowBase = blockIdx.x * TILE_M;
    const int colBase = blockIdx.y * TILE_N;

    v8f c0 = {}, c1 = {}, c2 = {}, c3 = {};

    for (int kc = 0; kc < Ktot; kc += TILE_K) {
        // ---- cooperative fill: A tile (zero-padded at edges) ----
#pragma unroll
        for (int i = 0; i < (TILE_M * TILE_K) / 256; ++i) {
            int idx = i * 256 + tid;
            int r = idx >> 6, cc = idx & 63;
            int grow = rowBase + r, gk = kc + cc;
            float v = 0.0f;
            if (grow < M && gk < Ktot)
                v = (gk < K0) ? A0[(long)grow * K0 + gk]
                              : A1[(long)grow * K1 + (gk - K0)];
            Alds[r * A_LDS_STRIDE + cc] = v;
        }
        // ---- cooperative fill: W tile ----
#pragma unroll
        for (int i = 0; i < (TILE_K * TILE_N) / 256; ++i) {
            int idx = i * 256 + tid;
            int r = idx >> 6, cc = idx & 63;
            int gk = kc + r, gc = colBase + cc;
            float v = 0.0f;
            if (gk < Ktot && gc < N) v = W[(long)gk * N + gc];
            Wlds[r * W_LDS_STRIDE + cc] = v;
        }
        __syncthreads();

        // ---- wave-level WMMA over this K chunk (no divergence: EXEC all-1s) ----
        const int rsel = (lane >> 4) << 1;     // 0 for lanes 0-15, 2 for lanes 16-31
        const int lcol = lane & 15;
        const float* Ab = &Alds[(wv * 16 + lcol) * A_LDS_STRIDE];
        for (int k = 0; k < TILE_K; k += 4) {
            v2f a;
            a.x = Ab[k + rsel];
            a.y = Ab[k + rsel + 1];
            const float* Wb0 = &Wlds[(k + rsel) * W_LDS_STRIDE + lcol];
            const float* Wb1 = Wb0 + W_LDS_STRIDE;
            v2f b0, b1, b2, b3;
            b0.x = Wb0[0];  b0.y = Wb1[0];
            b1.x = Wb0[16]; b1.y = Wb1[16];
            b2.x = Wb0[32]; b2.y = Wb1[32];
            b3.x = Wb0[48]; b3.y = Wb1[48];
            c0 = __builtin_amdgcn_wmma_f32_16x16x4_f32(false, a, false, b0, (short)0, c0, false, false);
            c1 = __builtin_amdgcn_wmma_f32_16x16x4_f32(false, a, false, b1, (short)0, c1, false, false);
            c2 = __builtin_amdgcn_wmma_f32_16x16x4_f32(false, a, false, b2, (short)0, c2, false, false);
            c3 = __builtin_amdgcn_wmma_f32_16x16x4_f32(false, a, false, b3, (short)0, c3, false, false);
        }
        __syncthreads();
    }

    const int lcol = lane & 15;
    const int row0 = rowBase + wv * 16 + ((lane >> 4) << 3);
    epilogue_store(c0, row0, colBase +  0 + lcol, M, N, addv, bias, do_relu, D, D2);
    epilogue_store(c1, row0, colBase + 16 + lcol, M, N, addv, bias, do_relu, D, D2);
    epilogue_store(c2, row0, colBase + 32 + lcol, M, N, addv, bias, do_relu, D, D2);
    epilogue_store(c3, row0, colBase + 48 + lcol, M, N, addv, bias, do_relu, D, D2);
}

// ---------------------------------------------------------------------------
// a_message[a] = sum_{j<6} message[a2b[a][j]]   (float4 over H)
// ---------------------------------------------------------------------------
__global__ void k_atom_sum(const float4* __restrict__ msg, const int* __restrict__ a2b,
                           float4* __restrict__ out, long Na) {
    long idx = (long)blockIdx.x * blockDim.x + threadIdx.x;
    long total = Na * HQ;
    if (idx >= total) return;
    long a = idx / HQ;
    int  h = (int)(idx % HQ);
    const int* nb = &a2b[a * MAXNB];
    float4 s = make_float4(0.f, 0.f, 0.f, 0.f);
#pragma unroll
    for (int j = 0; j < MAXNB; ++j) {
        float4 v = msg[(long)nb[j] * HQ + h];
        s.x += v.x; s.y += v.y; s.z += v.z; s.w += v.w;
    }
    out[idx] = s;
}

// ---------------------------------------------------------------------------
// m[b] = a_message[b2a[b]] - message[b2revb[b]]
// ---------------------------------------------------------------------------
__global__ void k_bond_delta(const float4* __restrict__ amsg, const float4* __restrict__ msg,
                             const int* __restrict__ b2a, const int* __restrict__ b2revb,
                             float4* __restrict__ out, long Nb) {
    long idx = (long)blockIdx.x * blockDim.x + threadIdx.x;
    long total = Nb * HQ;
    if (idx >= total) return;
    long b = idx / HQ;
    int  h = (int)(idx % HQ);
    float4 p = amsg[(long)b2a[b] * HQ + h];
    float4 q = msg[(long)b2revb[b] * HQ + h];
    out[idx] = make_float4(p.x - q.x, p.y - q.y, p.z - q.z, p.w - q.w);
}

// ---------------------------------------------------------------------------
// message_und[p] = 0.5*(message[d1[p]] + message[d2[p]]),  d1=2p, d2=(p?2p-1:0)
// ---------------------------------------------------------------------------
__global__ void k_pair(const float4* __restrict__ msg, float4* __restrict__ out, long Np) {
    long idx = (long)blockIdx.x * blockDim.x + threadIdx.x;
    long total = Np * HQ;
    if (idx >= total) return;
    long p = idx / HQ;
    int  h = (int)(idx % HQ);
    long d1 = 2 * p;
    long d2 = (p == 0) ? 0 : (2 * p - 1);
    float4 u = msg[d1 * HQ + h];
    float4 v = msg[d2 * HQ + h];
    out[idx] = make_float4(0.5f * (u.x + v.x), 0.5f * (u.y + v.y),
                           0.5f * (u.z + v.z), 0.5f * (u.w + v.w));
}

// ---------------------------------------------------------------------------
// BatchNorm stats over rows [1, Rtot): per-column sum / sumsq.
// blockDim = 320; thread c (<300) owns column c, block owns 256 rows. Coalesced.
// ---------------------------------------------------------------------------
__global__ void k_bn_stats(const float* __restrict__ X, float* __restrict__ sum,
                           float* __restrict__ sq, long Rtot) {
    int c = threadIdx.x;
    if (c >= HID) return;
    long r0 = 1 + (long)blockIdx.x * 256;
    float s = 0.f, s2 = 0.f;
    for (int i = 0; i < 256; ++i) {
        long r = r0 + i;
        if (r < Rtot) {
            float v = X[r * HID + c];
            s += v;
            s2 += v * v;
        }
    }
    atomicAdd(&sum[c], s);
    atomicAdd(&sq[c], s2);
}

__global__ void k_bn_apply(const float* __restrict__ X, const float* __restrict__ sum,
                           const float* __restrict__ sq, const float* __restrict__ gamma,
                           const float* __restrict__ beta, float* __restrict__ out, long Rtot) {
    long idx = (long)blockIdx.x * blockDim.x + threadIdx.x;
    long total = (Rtot - 1) * HID;
    if (idx >= total) return;
    int  c = (int)(idx % HID);
    long r = idx / HID + 1;
    float n   = (float)(Rtot - 1);
    float mu  = sum[c] / n;
    float var = sq[c] / n - mu * mu;
    float v   = X[r * HID + c];
    out[idx] = gamma[c] * (v - mu) * rsqrtf(var + BN_EPS) + beta[c];
}

__global__ void k_zero(float* __restrict__ p, long n) {
    long idx = (long)blockIdx.x * blockDim.x + threadIdx.x;
    if (idx < n) p[idx] = 0.0f;
}

// ---------------------------------------------------------------------------
extern "C" void kernel_launch(void* const* d_in, const int* in_sizes, int n_in,
                              void* d_out, int out_size, void* d_ws, size_t ws_size,
                              hipStream_t stream) {
    const float* f_atoms = (const float*)d_in[0];
    const float* f_bonds = (const float*)d_in[1];
    const int*   a2b     = (const int*)d_in[2];
    const int*   b2a     = (const int*)d_in[3];
    const int*   b2revb  = (const int*)d_in[4];
    const float* W_i     = (const float*)d_in[5];
    const float* W_h     = (const float*)d_in[6];
    const float* W_o     = (const float*)d_in[7];
    const float* b_o     = (const float*)d_in[8];
    const float* gamma   = (const float*)d_in[9];
    const float* beta    = (const float*)d_in[10];

    const long Na = in_sizes[0] / ATOMF;
    const long Nb = in_sizes[3];
    const long Np = (Nb + 1) / 2;

    // workspace layout (floats); und/hid overlay the dead mbuf region
    float* ws      = (float*)d_ws;
    float* inp     = ws;                       // [Nb, H]
    float* message = inp  + Nb * HID;          // [Nb, H]
    float* mbuf    = message + Nb * HID;       // [Nb, H]
    float* amsg    = mbuf + Nb * HID;          // [Na, H]
    float* und     = mbuf;                     // [Np, H]  (reuses mbuf)
    float* hid     = mbuf + Np * HID;          // [Na, H]  (reuses mbuf)
    float* stats   = amsg + Na * HID;          // 4*H: sum_a, sq_a, sum_b, sq_b
    float* sum_a = stats, *sq_a = stats + HID, *sum_b = stats + 2*HID, *sq_b = stats + 3*HID;

    const dim3 blk(256);
    const int  NBLK = (HID + TILE_N - 1) / TILE_N;  // 5

    // 1) inp = f_bonds @ W_i ; message = relu(inp)
    {
        dim3 grid((unsigned)((Nb + TILE_M - 1) / TILE_M), NBLK);
        k_gemm_wmma_f32<<<grid, blk, 0, stream>>>(f_bonds, BONDF, nullptr, 0, W_i,
                                                  nullptr, nullptr, message, inp,
                                                  (int)Nb, HID, 1);
    }

    // 2) message passing (DEPTH-1 = 2 iterations)
    for (int it = 0; it < 2; ++it) {
        long na_e = Na * HQ;
        k_atom_sum<<<dim3((unsigned)((na_e + 255) / 256)), blk, 0, stream>>>(
            (const float4*)message, a2b, (float4*)amsg, Na);
        long nb_e = Nb * HQ;
        k_bond_delta<<<dim3((unsigned)((nb_e + 255) / 256)), blk, 0, stream>>>(
            (const float4*)amsg, (const float4*)message, b2a, b2revb, (float4*)mbuf, Nb);
        dim3 grid((unsigned)((Nb + TILE_M - 1) / TILE_M), NBLK);
        k_gemm_wmma_f32<<<grid, blk, 0, stream>>>(mbuf, HID, nullptr, 0, W_h,
                                                  inp, nullptr, message, nullptr,
                                                  (int)Nb, HID, 1);
    }

    // 3) undirected pairing (into reused scratch)
    {
        long np_e = Np * HQ;
        k_pair<<<dim3((unsigned)((np_e + 255) / 256)), blk, 0, stream>>>(
            (const float4*)message, (float4*)und, Np);
    }

    // 4) final neighbor sum + output GEMM on concat(f_atoms, a_message)
    {
        long na_e = Na * HQ;
        k_atom_sum<<<dim3((unsigned)((na_e + 255) / 256)), blk, 0, stream>>>(
            (const float4*)message, a2b, (float4*)amsg, Na);
        dim3 grid((unsigned)((Na + TILE_M - 1) / TILE_M), NBLK);
        k_gemm_wmma_f32<<<grid, blk, 0, stream>>>(f_atoms, ATOMF, amsg, HID, W_o,
                                                  nullptr, b_o, hid, nullptr,
                                                  (int)Na, HID, 1);
    }

    // 5) BatchNorm (training-mode batch stats over rows 1:)
    k_zero<<<dim3(5), blk, 0, stream>>>(stats, 4 * HID);
    k_bn_stats<<<dim3((unsigned)((Na - 1 + 255) / 256)), dim3(320), 0, stream>>>(hid, sum_a, sq_a, Na);
    k_bn_stats<<<dim3((unsigned)((Np - 1 + 255) / 256)), dim3(320), 0, stream>>>(und, sum_b, sq_b, Np);

    float* out_atoms = (float*)d_out;
    float* out_bonds = out_atoms + (Na - 1) * HID;
    {
        long n1 = (Na - 1) * HID;
        k_bn_apply<<<dim3((unsigned)((n1 + 255) / 256)), blk, 0, stream>>>(
            hid, sum_a, sq_a, gamma, beta, out_atoms, Na);
        long n2 = (Np - 1) * HID;
        k_bn_apply<<<dim3((unsigned)((n2 + 255) / 256)), blk, 0, stream>>>(
            und, sum_b, sq_b, gamma, beta, out_bonds, Np);
    }
}